// PPI_GNN_Optimized_20856361189537
// MI455X (gfx1250) — compile-verified
//
#include <hip/hip_runtime.h>
#include <hip/hip_bf16.h>

// ---------------------------------------------------------------------------
// PPI GNN forward on gfx1250.
//  - Dense GEMMs use V_WMMA_F32_16X16X4_F32 (full fp32, wave32).
//    Each wave computes TWO 16 x (NT*16) output tiles; the weight panel for
//    the block's column group is staged in LDS once. B-fragments are loaded
//    into distinct registers before the WMMA group so ds_loads pipeline.
//  - Edge softmax/aggregation uses float atomics (segment max/sum/scatter).
// ---------------------------------------------------------------------------

typedef __attribute__((ext_vector_type(2))) float v2f;
typedef __attribute__((ext_vector_type(8))) float v8f;

#define N_NODES 50000
#define N_EDGES 800000
#define E_TOT   850000        // edges + self loops
#define N_PRED  250000
#define EF_K    68            // 66/67 padded to multiple of 4

__device__ __forceinline__ float leakyf(float v, float s) { return v > 0.f ? v : v * s; }

// ---------------- edge list with self loops ----------------
__global__ void k_build_edges(const int* __restrict__ ei, int* __restrict__ src,
                              int* __restrict__ dst) {
    int i = blockIdx.x * blockDim.x + threadIdx.x;
    if (i >= E_TOT) return;
    if (i < N_EDGES) { src[i] = ei[i]; dst[i] = ei[N_EDGES + i]; }
    else             { int n = i - N_EDGES; src[i] = n; dst[i] = n; }
}

__global__ void k_fill(float* __restrict__ p, float v, int n) {
    int i = blockIdx.x * blockDim.x + threadIdx.x;
    if (i < n) p[i] = v;
}

__global__ void k_rowbias(float* __restrict__ out, const float* __restrict__ bias,
                          int total, int D) {
    int i = blockIdx.x * blockDim.x + threadIdx.x;
    if (i < total) out[i] = bias[i % D];
}

// ---------------- fp32 WMMA GEMM: C[M,D] = A[M,K] @ W[D,K]^T + bias ----------
// Block: 256 threads (8 waves). Block owns 16 M-tiles x one (NT*16)-col group:
// wave w handles M-tiles (mg*16 + w) and (mg*16 + 8 + w).
// Per K-step each wave: 2 global b64 (A) + NT ds b64 (B) -> 2*NT WMMAs.
// A lane layout (16x4 f32): lanes 0-15 -> M=lane, K={k,k+1}; 16-31 -> K={k+2,k+3}.
// B lane layout (4x16 f32): lanes 0-15 -> N=lane, K={k,k+1}; 16-31 -> K={k+2,k+3}.
// C layout: VGPR r, lanes 0-15 -> (M=r, N=lane); lanes 16-31 -> (M=r+8, N=lane-16).
template <int NT>
__global__ void k_gemm_wmma(const float* __restrict__ A, const float* __restrict__ W,
                            const float* __restrict__ bias, float* __restrict__ C,
                            int M, int K, int D, float slope) {
    extern __shared__ float Bs[];                 // [NT*16][K]
    const int NW = NT * 16;
    int ngroups = D / NW;
    int ng = blockIdx.x % ngroups;
    int mg = blockIdx.x / ngroups;
    int colbase = ng * NW;

    // cooperative load of the weight panel (rows are K floats, K % 4 == 0)
    {
        const float4* Wp = (const float4*)(W + (size_t)colbase * K);
        float4*       Sp = (float4*)Bs;
        int total4 = (NW * K) >> 2;
        for (int i = threadIdx.x; i < total4; i += 256) Sp[i] = Wp[i];
    }
    __syncthreads();

    int wav  = threadIdx.x >> 5;
    int lane = threadIdx.x & 31;
    int Mt   = M >> 4;
    int mt0  = mg * 16 + wav;
    int mt1  = mg * 16 + 8 + wav;
    if (mt0 >= Mt) return;                        // whole-wave skip (after barrier)
    bool has1 = (mt1 < Mt);

    int hi = lane >> 4, l15 = lane & 15;
    const float* arow0 = A + (size_t)(mt0 * 16 + l15) * K + hi * 2;
    const float* arow1 = A + (size_t)((has1 ? mt1 : mt0) * 16 + l15) * K + hi * 2;
    const float* brow  = Bs + (size_t)l15 * K + hi * 2;

    v8f acc0[NT], acc1[NT];
#pragma unroll
    for (int j = 0; j < NT; ++j) {
        acc0[j] = (v8f){0.f,0.f,0.f,0.f,0.f,0.f,0.f,0.f};
        acc1[j] = (v8f){0.f,0.f,0.f,0.f,0.f,0.f,0.f,0.f};
    }

    for (int k = 0; k < K; k += 4) {
        // issue all loads first (distinct registers) so they pipeline
        v2f a0 = *(const v2f*)(arow0 + k);
        v2f a1 = *(const v2f*)(arow1 + k);
        v2f b[NT];
#pragma unroll
        for (int j = 0; j < NT; ++j)
            b[j] = *(const v2f*)(brow + (size_t)j * 16 * K + k);
#pragma unroll
        for (int j = 0; j < NT; ++j)
            acc0[j] = __builtin_amdgcn_wmma_f32_16x16x4_f32(
                false, a0, false, b[j], (short)0, acc0[j], false, false);
#pragma unroll
        for (int j = 0; j < NT; ++j)
            acc1[j] = __builtin_amdgcn_wmma_f32_16x16x4_f32(
                false, a1, false, b[j], (short)0, acc1[j], false, false);
    }

#pragma unroll
    for (int j = 0; j < NT; ++j) {
        int col = colbase + j * 16 + l15;
        float bv = bias[col];
#pragma unroll
        for (int r = 0; r < 8; ++r) {
            int row = mt0 * 16 + r + hi * 8;
            float v = acc0[j][r] + bv;
            if (slope >= 0.f) v = leakyf(v, slope);
            C[(size_t)row * D + col] = v;
        }
        if (has1) {
#pragma unroll
            for (int r = 0; r < 8; ++r) {
                int row = mt1 * 16 + r + hi * 8;
                float v = acc1[j][r] + bv;
                if (slope >= 0.f) v = leakyf(v, slope);
                C[(size_t)row * D + col] = v;
            }
        }
    }
}

// ---------------- GATv2 edge softmax ----------------
__device__ __forceinline__ void atomicMaxF(float* addr, float val) {
    if (val >= 0.f) atomicMax((int*)addr, __float_as_int(val));
    else            atomicMin((unsigned int*)addr, __float_as_uint(val));
}

__global__ void k_logit_max(const float* __restrict__ xl, const float* __restrict__ xr,
                            const float* __restrict__ att, const int* __restrict__ src,
                            const int* __restrict__ dst, float* __restrict__ logit,
                            float* __restrict__ m, int H, int C) {
    int t = blockIdx.x * blockDim.x + threadIdx.x;
    if (t >= E_TOT * H) return;
    int e = t / H, h = t - e * H;
    int s = src[e], d = dst[e];
    const float* pl = xl + ((size_t)s * H + h) * C;
    const float* pr = xr + ((size_t)d * H + h) * C;
    const float* pa = att + h * C;
    float acc = 0.f;
    for (int c = 0; c < C; ++c) acc += pa[c] * leakyf(pl[c] + pr[c], 0.2f);
    logit[t] = acc;
    atomicMaxF(&m[(size_t)d * H + h], acc);
}

__global__ void k_w_den(const float* __restrict__ logit, const int* __restrict__ dst,
                        const float* __restrict__ m, float* __restrict__ wbuf,
                        float* __restrict__ den, int H) {
    int t = blockIdx.x * blockDim.x + threadIdx.x;
    if (t >= E_TOT * H) return;
    int e = t / H, h = t - e * H;
    int d = dst[e];
    float w = __expf(logit[t] - m[(size_t)d * H + h]);
    wbuf[t] = w;
    atomicAdd(&den[(size_t)d * H + h], w);
}

__global__ void k_aggregate(const float* __restrict__ xl, const float* __restrict__ wbuf,
                            const float* __restrict__ den, const int* __restrict__ src,
                            const int* __restrict__ dst, float* __restrict__ out,
                            int H, int C) {
    int t = blockIdx.x * blockDim.x + threadIdx.x;
    if (t >= E_TOT * H) return;
    int e = t / H, h = t - e * H;
    int s = src[e], d = dst[e];
    float alpha = wbuf[t] / den[(size_t)d * H + h];
    const float* pl = xl + ((size_t)s * H + h) * C;
    float* po = out + ((size_t)d * H + h) * C;
    for (int c = 0; c < C; ++c) atomicAdd(&po[c], alpha * pl[c]);
}

// ---------------- BN (inference) + leaky 0.1 ----------------
__global__ void k_bn_leaky(const float* __restrict__ x, const float* __restrict__ gamma,
                           const float* __restrict__ beta, const float* __restrict__ mean,
                           const float* __restrict__ var, float* __restrict__ out,
                           int total, int D) {
    int i = blockIdx.x * blockDim.x + threadIdx.x;
    if (i >= total) return;
    int j = i % D;
    float v = (x[i] - mean[j]) * rsqrtf(var[j] + 1e-5f) * gamma[j] + beta[j];
    out[i] = leakyf(v, 0.1f);
}

__global__ void k_add(float* __restrict__ a, const float* __restrict__ b, int n) {
    int i = blockIdx.x * blockDim.x + threadIdx.x;
    if (i < n) a[i] += b[i];
}

// ---------------- edge-feature build (padded to 68 cols) ----------------
__global__ void k_build_ef(const float* __restrict__ z, const int* __restrict__ ps,
                           const int* __restrict__ pd, const float* __restrict__ pert,
                           const float* __restrict__ pr, float* __restrict__ ef) {
    int i = blockIdx.x * blockDim.x + threadIdx.x;
    if (i >= N_PRED * EF_K) return;
    int p = i / EF_K, c = i - p * EF_K;
    int a = ps[p], b = pd[p];
    float v;
    if (c < 32)       v = z[(size_t)a * 32 + c];
    else if (c < 64)  v = z[(size_t)b * 32 + (c - 32)];
    else if (c == 64) v = pr[a];
    else if (c == 65) v = pr[b];
    else if (c == 66) v = pert[a] * pert[b];
    else              v = 0.f;
    ef[i] = v;
}

__global__ void k_pad_w(const float* __restrict__ W, float* __restrict__ Wp,
                        int D, int Kin, int Kout) {
    int i = blockIdx.x * blockDim.x + threadIdx.x;
    if (i >= D * Kout) return;
    int d = i / Kout, k = i - d * Kout;
    Wp[i] = (k < Kin) ? W[(size_t)d * Kin + k] : 0.f;
}

// ---------------- final 64->2 layer ----------------
__global__ void k_final2(const float* __restrict__ t2, const float* __restrict__ W,
                         const float* __restrict__ b, float* __restrict__ out) {
    int p = blockIdx.x * blockDim.x + threadIdx.x;
    if (p >= N_PRED) return;
    const float* tp = t2 + (size_t)p * 64;
    float a0 = b[0], a1 = b[1];
    for (int c = 0; c < 64; ++c) {
        float v = tp[c];
        a0 += W[c] * v;
        a1 += W[64 + c] * v;
    }
    out[(size_t)p * 2 + 0] = a0;
    out[(size_t)p * 2 + 1] = a1;
}

// ---------------------------------------------------------------------------
static inline int cdiv(long long a, long long b) { return (int)((a + b - 1) / b); }

// D % 64 == 0 -> NT=4 (two 16x64 tiles per wave); else (D==32) -> NT=2.
static void launch_gemm(const float* A, const float* W, const float* bias, float* C,
                        int M, int K, int D, float slope, hipStream_t stream) {
    int mtiles  = M >> 4;
    int mgroups = cdiv(mtiles, 16);               // 16 M-tiles per block
    if (D % 64 == 0) {
        int ngroups = D / 64;
        size_t smem = (size_t)64 * K * 4;
        k_gemm_wmma<4><<<mgroups * ngroups, 256, smem, stream>>>(A, W, bias, C, M, K, D, slope);
    } else {
        int ngroups = D / 32;
        size_t smem = (size_t)32 * K * 4;
        k_gemm_wmma<2><<<mgroups * ngroups, 256, smem, stream>>>(A, W, bias, C, M, K, D, slope);
    }
}

static void run_gat(const float* x, int inD, int H, int C,
                    const float* Wl, const float* bl, const float* Wr, const float* br,
                    const float* att, const float* bias,
                    const int* src, const int* dst,
                    float* xl, float* xr, float* go, float* logit, float* wbuf,
                    float* m, float* den, hipStream_t stream) {
    int D = H * C;
    launch_gemm(x, Wl, bl, xl, N_NODES, inD, D, -1.f, stream);
    launch_gemm(x, Wr, br, xr, N_NODES, inD, D, -1.f, stream);
    k_fill<<<cdiv(N_NODES * H, 256), 256, 0, stream>>>(m, -3.0e38f, N_NODES * H);
    k_fill<<<cdiv(N_NODES * H, 256), 256, 0, stream>>>(den, 0.f, N_NODES * H);
    k_rowbias<<<cdiv((long long)N_NODES * D, 256), 256, 0, stream>>>(go, bias, N_NODES * D, D);
    int EH = E_TOT * H;
    k_logit_max<<<cdiv(EH, 256), 256, 0, stream>>>(xl, xr, att, src, dst, logit, m, H, C);
    k_w_den<<<cdiv(EH, 256), 256, 0, stream>>>(logit, dst, m, wbuf, den, H);
    k_aggregate<<<cdiv(EH, 256), 256, 0, stream>>>(xl, wbuf, den, src, dst, go, H, C);
}

extern "C" void kernel_launch(void* const* d_in, const int* in_sizes, int n_in,
                              void* d_out, int out_size, void* d_ws, size_t ws_size,
                              hipStream_t stream) {
    (void)in_sizes; (void)n_in; (void)out_size; (void)ws_size;

    // ---- inputs (pytree flatten: top-level insertion order, dict keys sorted) ----
    const float* x    = (const float*)d_in[0];
    const int*   ei   = (const int*)  d_in[1];   // [2, N_EDGES]
    const int*   pe   = (const int*)  d_in[2];   // [2, N_PRED]
    const float* pert = (const float*)d_in[3];
    const float* pr   = (const float*)d_in[4];
    const float* bn1_beta = (const float*)d_in[5];
    const float* bn1_gam  = (const float*)d_in[6];
    const float* bn1_mean = (const float*)d_in[7];
    const float* bn1_var  = (const float*)d_in[8];
    const float* bn2_beta = (const float*)d_in[9];
    const float* bn2_gam  = (const float*)d_in[10];
    const float* bn2_mean = (const float*)d_in[11];
    const float* bn2_var  = (const float*)d_in[12];
    const float* bn3_beta = (const float*)d_in[13];
    const float* bn3_gam  = (const float*)d_in[14];
    const float* bn3_mean = (const float*)d_in[15];
    const float* bn3_var  = (const float*)d_in[16];
    const float* c1_att = (const float*)d_in[17];
    const float* c1_bias= (const float*)d_in[18];
    const float* c1_Wl  = (const float*)d_in[19];
    const float* c1_bl  = (const float*)d_in[20];
    const float* c1_Wr  = (const float*)d_in[21];
    const float* c1_br  = (const float*)d_in[22];
    const float* c2_att = (const float*)d_in[23];
    const float* c2_bias= (const float*)d_in[24];
    const float* c2_Wl  = (const float*)d_in[25];
    const float* c2_bl  = (const float*)d_in[26];
    const float* c2_Wr  = (const float*)d_in[27];
    const float* c2_br  = (const float*)d_in[28];
    const float* c3_att = (const float*)d_in[29];
    const float* c3_bias= (const float*)d_in[30];
    const float* c3_Wl  = (const float*)d_in[31];
    const float* c3_bl  = (const float*)d_in[32];
    const float* c3_Wr  = (const float*)d_in[33];
    const float* c3_br  = (const float*)d_in[34];
    const float* ep0_W = (const float*)d_in[35];  // [128,66]
    const float* ep0_b = (const float*)d_in[36];
    const float* ep1_W = (const float*)d_in[37];  // [64,128]
    const float* ep1_b = (const float*)d_in[38];
    const float* ep2_W = (const float*)d_in[39];  // [2,64]
    const float* ep2_b = (const float*)d_in[40];
    const float* pp0_W = (const float*)d_in[41];  // [128,67]
    const float* pp0_b = (const float*)d_in[42];
    const float* pp1_W = (const float*)d_in[43];
    const float* pp1_b = (const float*)d_in[44];
    const float* pp2_W = (const float*)d_in[45];
    const float* pp2_b = (const float*)d_in[46];
    const float* res_W = (const float*)d_in[47];  // [32,256]
    const float* res_b = (const float*)d_in[48];

    float* out = (float*)d_out;   // [250000,2] normal ++ [250000,2] perturbed

    // ---- workspace layout ----
    int* src = (int*)d_ws;
    int* dst = src + E_TOT;
    float* F = (float*)((char*)d_ws + (size_t)2 * E_TOT * 4);   // 8-byte aligned
    // phase 1 (GNN)
    float* XL = F +  0LL;          // N*256
    float* XR = F + 12800000LL;    // N*256
    float* GO = F + 25600000LL;    // N*256
    float* H1 = F + 38400000LL;    // N*256
    float* H2 = F + 51200000LL;    // N*128
    float* LG = F + 57600000LL;    // E_TOT*4
    float* WWb= F + 61000000LL;    // E_TOT*4
    float* MM = F + 64400000LL;    // N*4
    float* DN = F + 64600000LL;    // N*4
    float* Z  = F + 65000000LL;    // N*32  (survives into phase 2)
    float* ZR = F + 66600000LL;    // N*32
    float* WP = F + 68200000LL;    // 128*68
    // phase 2 (edge MLPs) — aliases the dead GNN buffers
    float* EF = F +  0LL;          // P*68
    float* T1 = F + 17000000LL;    // P*128
    float* T2 = F + 49000000LL;    // P*64   (ends at 65e6 <= Z)

    // ---- build edge list with self loops ----
    k_build_edges<<<cdiv(E_TOT, 256), 256, 0, stream>>>(ei, src, dst);

    // ---- layer 1: GATv2(128 -> 4x64) + BN + leaky ----
    run_gat(x, 128, 4, 64, c1_Wl, c1_bl, c1_Wr, c1_br, c1_att, c1_bias,
            src, dst, XL, XR, GO, LG, WWb, MM, DN, stream);
    k_bn_leaky<<<cdiv((long long)N_NODES * 256, 256), 256, 0, stream>>>(
        GO, bn1_gam, bn1_beta, bn1_mean, bn1_var, H1, N_NODES * 256, 256);

    // ---- layer 2: GATv2(256 -> 4x32) + BN + leaky ----
    run_gat(H1, 256, 4, 32, c2_Wl, c2_bl, c2_Wr, c2_br, c2_att, c2_bias,
            src, dst, XL, XR, GO, LG, WWb, MM, DN, stream);
    k_bn_leaky<<<cdiv((long long)N_NODES * 128, 256), 256, 0, stream>>>(
        GO, bn2_gam, bn2_beta, bn2_mean, bn2_var, H2, N_NODES * 128, 128);

    // ---- layer 3: GATv2(128 -> 1x32) + BN + leaky -> Z ----
    run_gat(H2, 128, 1, 32, c3_Wl, c3_bl, c3_Wr, c3_br, c3_att, c3_bias,
            src, dst, XL, XR, GO, LG, WWb, MM, DN, stream);
    k_bn_leaky<<<cdiv((long long)N_NODES * 32, 256), 256, 0, stream>>>(
        GO, bn3_gam, bn3_beta, bn3_mean, bn3_var, Z, N_NODES * 32, 32);

    // ---- residual: Z += H1 @ res_W^T + res_b ----
    launch_gemm(H1, res_W, res_b, ZR, N_NODES, 256, 32, -1.f, stream);
    k_add<<<cdiv(N_NODES * 32, 256), 256, 0, stream>>>(Z, ZR, N_NODES * 32);

    // ---- edge features (shared by both predictors; col 66 = pert, 67 = 0) ----
    const int* ps = pe;
    const int* pd = pe + N_PRED;
    k_build_ef<<<cdiv((long long)N_PRED * EF_K, 256), 256, 0, stream>>>(Z, ps, pd, pert, pr, EF);

    // ---- normal edge predictor: 66->128->64->2 (pad W0 to K=68) ----
    k_pad_w<<<cdiv(128 * EF_K, 256), 256, 0, stream>>>(ep0_W, WP, 128, 66, EF_K);
    launch_gemm(EF, WP,   ep0_b, T1, N_PRED, EF_K, 128, 0.1f, stream);
    launch_gemm(T1, ep1_W, ep1_b, T2, N_PRED, 128,  64,  0.1f, stream);
    k_final2<<<cdiv(N_PRED, 256), 256, 0, stream>>>(T2, ep2_W, ep2_b, out);

    // ---- perturbed edge predictor: 67->128->64->2 ----
    k_pad_w<<<cdiv(128 * EF_K, 256), 256, 0, stream>>>(pp0_W, WP, 128, 67, EF_K);
    launch_gemm(EF, WP,   pp0_b, T1, N_PRED, EF_K, 128, 0.1f, stream);
    launch_gemm(T1, pp1_W, pp1_b, T2, N_PRED, 128,  64,  0.1f, stream);
    k_final2<<<cdiv(N_PRED, 256), 256, 0, stream>>>(T2, pp2_W, pp2_b, out + (size_t)2 * N_PRED);
}